// RoIFPPool3d_46084999086897
// MI455X (gfx1250) — compile-verified
//
#include <hip/hip_runtime.h>

typedef __attribute__((ext_vector_type(2))) float v2f;
typedef __attribute__((ext_vector_type(8))) float v8f;

#define B_   2
#define N_   8192
#define C_   128
#define R_   128
#define G_   125
#define M_   16000      // R_*G_
#define BN_EPS 1e-5f

// ---------------------------------------------------------------------------
// K1: build grid points from rois.  grid[(b*M_+m)*3 + d]
// ---------------------------------------------------------------------------
__global__ void grid_kernel(const float* __restrict__ rois,
                            float* __restrict__ grid) {
    int t = blockIdx.x * blockDim.x + threadIdx.x;
    if (t >= B_ * M_) return;
    int b = t / M_;
    int m = t - b * M_;
    int r = m / G_;
    int g = m - r * G_;
    int ix = g / 25;
    int rem = g - ix * 25;
    int iy = rem / 5;
    int iz = rem - iy * 5;
    const float* roi = rois + (b * R_ + r) * 7;
    float cx = roi[0], cy = roi[1], cz = roi[2] + 0.5f * roi[5];
    float dx = roi[3], dy = roi[4], dz = roi[5];
    float bx = (float)ix * 0.2f - 0.4f;
    float by = (float)iy * 0.2f - 0.4f;
    float bz = (float)iz * 0.2f - 0.4f;
    float* o = grid + t * 3;
    o[0] = bx * dx + cx;
    o[1] = by * dy + cy;
    o[2] = bz * dz + cz;
}

// ---------------------------------------------------------------------------
// CDNA5 async global->LDS copy (16B per lane), tracked by ASYNCcnt.
// GV mode: vdst = per-lane LDS byte offset, vaddr = per-lane 64-bit address.
// ---------------------------------------------------------------------------
__device__ __forceinline__ void async_copy_b128(unsigned lds_off,
                                                const void* gaddr) {
    asm volatile("global_load_async_to_lds_b128 %0, %1, off"
                 :
                 : "v"(lds_off), "v"(gaddr)
                 : "memory");
}

// ---------------------------------------------------------------------------
// K2: 3-NN over 8192 points per query; pts double-buffered through LDS with
// CDNA5 async-to-LDS loads. One block = 128 queries of one batch.
// Tile = 512 points = 6144 B = 3 x (128 lanes x 16 B) async issues.
// ---------------------------------------------------------------------------
#define KT  512
#define KTF (KT * 3)            // floats per tile
#define KTB (KT * 12)           // bytes per tile (6144)
__global__ void __launch_bounds__(128) knn_kernel(
        const float* __restrict__ grid, const float* __restrict__ pts,
        int* __restrict__ nidx, float* __restrict__ nw) {
    __shared__ float sp[2][KTF];
    int tid = threadIdx.x;
    int b = blockIdx.x / 125;
    int m = (blockIdx.x % 125) * 128 + tid;
    const float* q = grid + (b * M_ + m) * 3;
    float gx = q[0], gy = q[1], gz = q[2];
    float d0 = 3.4e38f, d1 = 3.4e38f, d2 = 3.4e38f;
    int   i0 = 0, i1 = 0, i2 = 0;

    const char* gb = (const char*)(pts + (size_t)b * N_ * 3);
    unsigned lds0 = (unsigned)(size_t)(void*)&sp[0][0];
    unsigned lds1 = (unsigned)(size_t)(void*)&sp[1][0];
    unsigned loff = (unsigned)tid * 16u;

    const int NTILES = N_ / KT;   // 16

    // prologue: tile 0 -> buffer 0
    async_copy_b128(lds0 + loff,        gb + loff);
    async_copy_b128(lds0 + loff + 2048, gb + loff + 2048);
    async_copy_b128(lds0 + loff + 4096, gb + loff + 4096);

    for (int t = 0; t < NTILES; ++t) {
        const float* cur = sp[t & 1];
        if (t + 1 < NTILES) {
            unsigned nb = (t & 1) ? lds0 : lds1;
            const char* g = gb + (size_t)(t + 1) * KTB;
            async_copy_b128(nb + loff,        g + loff);
            async_copy_b128(nb + loff + 2048, g + loff + 2048);
            async_copy_b128(nb + loff + 4096, g + loff + 4096);
            // in-order completion: <=3 outstanding => current tile landed
            asm volatile("s_wait_asynccnt 0x3" ::: "memory");
        } else {
            asm volatile("s_wait_asynccnt 0x0" ::: "memory");
        }
        __syncthreads();
        int n0 = t * KT;
        #pragma unroll 4
        for (int jj = 0; jj < KT; ++jj) {
            float px = cur[jj * 3 + 0];
            float py = cur[jj * 3 + 1];
            float pz = cur[jj * 3 + 2];
            float ddx = gx - px, ddy = gy - py, ddz = gz - pz;
            float d = ddx * ddx + ddy * ddy + ddz * ddz;
            int n = n0 + jj;
            if (d < d2) {
                if (d < d1) {
                    if (d < d0) { d2 = d1; i2 = i1; d1 = d0; i1 = i0; d0 = d; i0 = n; }
                    else        { d2 = d1; i2 = i1; d1 = d;  i1 = n; }
                } else          { d2 = d;  i2 = n; }
            }
        }
        __syncthreads();
    }
    float w0 = 1.0f / (d0 + 1e-8f);
    float w1 = 1.0f / (d1 + 1e-8f);
    float w2 = 1.0f / (d2 + 1e-8f);
    float inv = 1.0f / (w0 + w1 + w2);
    int base = (b * M_ + m) * 3;
    nidx[base + 0] = i0; nidx[base + 1] = i1; nidx[base + 2] = i2;
    nw[base + 0] = w0 * inv; nw[base + 1] = w1 * inv; nw[base + 2] = w2 * inv;
}

// ---------------------------------------------------------------------------
// K3: fused interp-gather + GEMM (WMMA f32 16x16x4) + BN + ReLU + permute.
// Block = 256 threads (8 wave32s), one 128(o) x 64(m) output tile.
// LDS interp tile pitch 72: rows k and k+2 (the two half-wave reads of a
// B fragment) differ by 144 dwords = 16 mod 64 banks -> disjoint bank sets.
// ---------------------------------------------------------------------------
#define BM      64
#define SIPITCH 72
__global__ void __launch_bounds__(256) gemm_kernel(
        const float* __restrict__ convw, const float* __restrict__ feat,
        const int* __restrict__ nidx, const float* __restrict__ nw,
        const float* __restrict__ gamma, const float* __restrict__ beta,
        const float* __restrict__ mean, const float* __restrict__ var,
        float* __restrict__ out) {
    __shared__ float sI[C_ * SIPITCH];     // interp tile [c][ml]
    __shared__ float sScale[C_], sShift[C_];
    __shared__ int   sIdx[BM * 3];
    __shared__ float sW3[BM * 3];

    int tid = threadIdx.x;
    int b   = blockIdx.x / 250;
    int m0  = (blockIdx.x % 250) * BM;

    if (tid < C_) {
        float sc = gamma[tid] * rsqrtf(var[tid] + BN_EPS);
        sScale[tid] = sc;
        sShift[tid] = beta[tid] - mean[tid] * sc;
    }
    if (tid < BM * 3) {
        int gbase = (b * M_ + m0) * 3;
        sIdx[tid] = nidx[gbase + tid];
        sW3[tid]  = nw[gbase + tid];
    }
    __syncthreads();

    // Build interp tile I[c][ml] = sum_k feat[b][c][idx] * w  (all L2-resident)
    const float* fb = feat + (size_t)b * C_ * N_;
    for (int t = tid; t < C_ * BM; t += 256) {
        int c  = t >> 6;
        int ml = t & 63;
        const float* f = fb + c * N_;
        int p = ml * 3;
        sI[c * SIPITCH + ml] = f[sIdx[p + 0]] * sW3[p + 0] +
                               f[sIdx[p + 1]] * sW3[p + 1] +
                               f[sIdx[p + 2]] * sW3[p + 2];
    }
    __syncthreads();

    int wave = tid >> 5;
    int lane = tid & 31;
    int o0   = wave * 16;          // this wave's 16-row output strip
    int ln   = lane & 15;
    int ksh  = (lane < 16) ? 0 : 2;   // ISA 16x4 fp32 A/B layout: K=0,1 | K=2,3
    int oA   = o0 + ln;

    // Preload all A fragments (W[o][k]) for K=128: 32 x float2 per lane.
    v2f afrag[32];
    #pragma unroll
    for (int kk = 0; kk < 32; ++kk) {
        const float* wp = convw + oA * C_ + kk * 4 + ksh;
        afrag[kk].x = wp[0];
        afrag[kk].y = wp[1];
    }

    for (int mt = 0; mt < 4; ++mt) {
        int mtb = mt * 16;
        v8f acc = {};
        #pragma unroll
        for (int kk = 0; kk < 32; ++kk) {
            int k0 = kk * 4 + ksh;
            v2f bf;
            bf.x = sI[(k0    ) * SIPITCH + mtb + ln];
            bf.y = sI[(k0 + 1) * SIPITCH + mtb + ln];
            acc = __builtin_amdgcn_wmma_f32_16x16x4_f32(
                      /*neg_a=*/false, afrag[kk],
                      /*neg_b=*/false, bf,
                      /*c_mod=*/(short)0, acc,
                      /*reuse_a=*/false, /*reuse_b=*/false);
        }
        // C/D layout: VGPR r -> row r (lanes 0-15) / row r+8 (lanes 16-31)
        int rowoff = (lane < 16) ? 0 : 8;
        int m   = m0 + mtb + ln;
        int roi = m / G_;
        int g   = m - roi * G_;
        #pragma unroll
        for (int r = 0; r < 8; ++r) {
            int o = o0 + r + rowoff;
            float y = acc[r] * sScale[o] + sShift[o];
            y = fmaxf(y, 0.0f);
            out[((size_t)(b * R_ + roi) * C_ + o) * G_ + g] = y;
        }
    }
}

// ---------------------------------------------------------------------------
extern "C" void kernel_launch(void* const* d_in, const int* in_sizes, int n_in,
                              void* d_out, int out_size, void* d_ws, size_t ws_size,
                              hipStream_t stream) {
    const float* pts   = (const float*)d_in[0];   // (B, N, 3)
    const float* feat  = (const float*)d_in[1];   // (B, C, N)
    const float* rois  = (const float*)d_in[2];   // (B, R, 7)
    const float* convw = (const float*)d_in[3];   // (C, C)
    const float* gamma = (const float*)d_in[4];
    const float* beta  = (const float*)d_in[5];
    const float* mean  = (const float*)d_in[6];
    const float* var   = (const float*)d_in[7];
    float* out = (float*)d_out;

    float* ws   = (float*)d_ws;
    float* grid = ws;                                  // B*M*3 floats
    int*   nidx = (int*)(ws + (size_t)B_ * M_ * 3);    // B*M*3 ints
    float* nw   = ws + 2 * (size_t)B_ * M_ * 3;        // B*M*3 floats

    grid_kernel<<<(B_ * M_ + 255) / 256, 256, 0, stream>>>(rois, grid);
    knn_kernel<<<B_ * 125, 128, 0, stream>>>(grid, pts, nidx, nw);
    gemm_kernel<<<B_ * 250, 256, 0, stream>>>(convw, feat, nidx, nw,
                                              gamma, beta, mean, var, out);
}